// SFGCN_34969623724073
// MI455X (gfx1250) — compile-verified
//
#include <hip/hip_runtime.h>
#include <math.h>

typedef __attribute__((ext_vector_type(2))) float v2f;
typedef __attribute__((ext_vector_type(8))) float v8f;

#define HDIM 128
#define NB 64

// ---------------- GEMM: S[M,128] = X[M,128] @ W[128,128] + bias ----------------
// fp32 WMMA 16x16x4. Block = 256 threads (8 waves); each wave owns a 16-col slab
// of W (all K in registers: 32 v2f frags) and computes 4 M-tiles (64 rows/block).
__global__ void gemm_bias_f32(const float* __restrict__ X, const float* __restrict__ W,
                              const float* __restrict__ bias, float* __restrict__ S, int M) {
  const int lane = threadIdx.x & 31;
  const int wave = threadIdx.x >> 5;
  const int n0 = wave * 16;              // 8 waves cover N = 128
  const int mbase = blockIdx.x * 64;
  const int lm = lane & 15;              // row (A) / col (B) within tile
  const int lk = (lane >> 4) * 2;        // K sub-offset per half-wave

  // B frags: B is 4x16 (KxN); VGPR0 holds K=lk, VGPR1 holds K=lk+1
  v2f bf[32];
#pragma unroll
  for (int kk = 0; kk < 32; ++kk) {
    const int k = kk * 4 + lk;
    bf[kk].x = W[(k + 0) * HDIM + n0 + lm];
    bf[kk].y = W[(k + 1) * HDIM + n0 + lm];
  }
  const float bv = bias[n0 + lm];

  for (int t = 0; t < 4; ++t) {
    const int m0 = mbase + t * 16;
    const float* xrow = X + (size_t)(m0 + lm) * HDIM + lk;
    v8f acc = {};
#pragma unroll
    for (int kk = 0; kk < 32; ++kk) {
      v2f a;                              // A is 16x4: lane holds row lm, K=lk,lk+1
      a.x = xrow[kk * 4 + 0];
      a.y = xrow[kk * 4 + 1];
      acc = __builtin_amdgcn_wmma_f32_16x16x4_f32(false, a, false, bf[kk],
                                                  (short)0, acc, false, false);
    }
    // C/D layout: VGPR r = row (r + 8*hi), col = lane&15
    const int hi = lane >> 4;
    const int n = n0 + lm;
#pragma unroll
    for (int r = 0; r < 8; ++r)
      S[(size_t)(m0 + r + 8 * hi) * HDIM + n] = acc[r] + bv;
  }
}

// ---------------- Edge scatter-add: H[dst] += S[src] * mask ----------------
__global__ void edge_scatter(const float* __restrict__ S, const int* __restrict__ src,
                             const int* __restrict__ dst, const float* __restrict__ mask,
                             float* __restrict__ H, int E) {
  const int lane = threadIdx.x & 31;
  const int e = blockIdx.x * 8 + (threadIdx.x >> 5);
  if (e >= E) return;
  const float m = mask ? mask[e] : 1.0f;
  if (m == 0.0f) return;                  // wave-uniform skip of dropped edges
  const int s = src[e], d = dst[e];
  float4 v = ((const float4*)(S + (size_t)s * HDIM))[lane];
  float* hp = H + (size_t)d * HDIM + lane * 4;
  atomicAdd(hp + 0, v.x * m);
  atomicAdd(hp + 1, v.y * m);
  atomicAdd(hp + 2, v.z * m);
  atomicAdd(hp + 3, v.w * m);
}

__global__ void relu_inplace(float* X, int n) {
  const int i = blockIdx.x * blockDim.x + threadIdx.x;
  if (i < n) X[i] = fmaxf(X[i], 0.0f);
}

__device__ inline float wave_sum(float v) {
#pragma unroll
  for (int o = 16; o > 0; o >>= 1) v += __shfl_xor(v, o, 32);
  return v;
}

// score = tanh((h . pw) / ||pw||), one wave per row
__global__ void node_score(const float* __restrict__ H, const float* __restrict__ pw,
                           float* __restrict__ score, int M) {
  const int lane = threadIdx.x & 31;
  const int r = blockIdx.x * 8 + (threadIdx.x >> 5);
  if (r >= M) return;
  const float4 w4 = ((const float4*)pw)[lane];
  const float4 h4 = ((const float4*)(H + (size_t)r * HDIM))[lane];
  float d  = h4.x * w4.x + h4.y * w4.y + h4.z * w4.z + h4.w * w4.w;
  float ww = w4.x * w4.x + w4.y * w4.y + w4.z * w4.z + w4.w * w4.w;
  d = wave_sum(d);
  ww = wave_sum(ww);
  if (lane == 0) score[r] = tanhf(d / sqrtf(ww));
}

// Per-graph bitonic top-k: block = n threads, descending by (score, -idx)
__global__ void topk_select(const float* __restrict__ score, int n, int k,
                            int* __restrict__ perm, int* __restrict__ mapping,
                            float* __restrict__ kept) {
  __shared__ float sc[512];
  __shared__ int si[512];
  const int g = blockIdx.x;
  const int t = threadIdx.x;              // blockDim.x == n (power of 2, <= 512)
  sc[t] = score[g * n + t];
  si[t] = t;
  __syncthreads();
  for (int ks = 2; ks <= n; ks <<= 1) {
    for (int j = ks >> 1; j > 0; j >>= 1) {
      const int ixj = t ^ j;
      if (ixj > t) {
        const bool descend = ((t & ks) == 0);
        const float a = sc[t], b = sc[ixj];
        const int ia = si[t], ib = si[ixj];
        const bool a_first = (a > b) || (a == b && ia < ib);  // descending criterion
        if (descend ? !a_first : a_first) {
          sc[t] = b; sc[ixj] = a; si[t] = ib; si[ixj] = ia;
        }
      }
      __syncthreads();
    }
  }
  if (t < k) {
    const int gnode = g * n + si[t];
    perm[g * k + t] = gnode;
    mapping[gnode] = g * k + t;
    kept[gnode] = 1.0f;
  }
}

// x_new[r] = H[perm[r]] * score[perm[r]], one wave per row
__global__ void gather_scale(const float* __restrict__ H, const float* __restrict__ score,
                             const int* __restrict__ perm, float* __restrict__ Xn, int rows) {
  const int lane = threadIdx.x & 31;
  const int r = blockIdx.x * 8 + (threadIdx.x >> 5);
  if (r >= rows) return;
  const int g = perm[r];
  const float s = score[g];
  float4 v = ((const float4*)(H + (size_t)g * HDIM))[lane];
  v.x *= s; v.y *= s; v.z *= s; v.w *= s;
  ((float4*)(Xn + (size_t)r * HDIM))[lane] = v;
}

// remap edges through pooling (safe in place: each thread reads its slot first)
__global__ void edge_remap(const int* src, const int* dst, const float* mask,
                           const int* mapping, const float* kept,
                           int* nsrc, int* ndst, float* nmask, int E) {
  const int e = blockIdx.x * blockDim.x + threadIdx.x;
  if (e >= E) return;
  const int s = src[e], d = dst[e];
  const float m = mask ? mask[e] : 1.0f;
  nmask[e] = m * kept[s] * kept[d];
  nsrc[e] = mapping[s];
  ndst[e] = mapping[d];
}

// readout: emb[g] += concat(max over rows, mean over rows); block = 128 (one feat/thread)
__global__ void readout_acc(const float* __restrict__ X, float* __restrict__ emb, int k) {
  const int g = blockIdx.x, c = threadIdx.x;
  float mx = -3.4e38f, sm = 0.0f;
  const float* p = X + (size_t)g * k * HDIM + c;
  for (int r = 0; r < k; ++r) {
    const float v = p[(size_t)r * HDIM];
    mx = fmaxf(mx, v);
    sm += v;
  }
  emb[g * 256 + c] += mx;
  emb[g * 256 + 128 + c] += sm / (float)k;
}

// attention over z = [emb1, emb2, (com1+com2)/2] + assemble all outputs
__global__ void finalize_att(const float* e1, const float* c1, const float* c2,
                             const float* e2, const float* W1, const float* b1,
                             const float* w2, float* out) {
  const int g = blockIdx.x, t = threadIdx.x;   // 128 threads
  __shared__ float z[3][256];
  __shared__ float red[128];
  __shared__ float sbeta[3];
  for (int i = t; i < 256; i += 128) {
    z[0][i] = e1[g * 256 + i];
    z[1][i] = e2[g * 256 + i];
    z[2][i] = 0.5f * (c1[g * 256 + i] + c2[g * 256 + i]);
  }
  __syncthreads();
  for (int j = 0; j < 3; ++j) {
    float acc = b1[t];
    for (int i = 0; i < 256; ++i) acc += z[j][i] * W1[i * 128 + t];
    red[t] = tanhf(acc) * w2[t];
    __syncthreads();
    for (int s = 64; s > 0; s >>= 1) {
      if (t < s) red[t] += red[t + s];
      __syncthreads();
    }
    if (t == 0) sbeta[j] = red[0];
    __syncthreads();
  }
  if (t == 0) {
    const float m = fmaxf(sbeta[0], fmaxf(sbeta[1], sbeta[2]));
    const float s0 = expf(sbeta[0] - m), s1 = expf(sbeta[1] - m), s2 = expf(sbeta[2] - m);
    const float inv = 1.0f / (s0 + s1 + s2);
    sbeta[0] = s0 * inv; sbeta[1] = s1 * inv; sbeta[2] = s2 * inv;
  }
  __syncthreads();
  const int OUT0 = 0, BETA = 16384, EMB1 = 16576, COM1 = 32960,
            COM2 = 49344, EMB2 = 65728, OUT2 = 82112;
  for (int i = t; i < 256; i += 128) {
    const float o = sbeta[0] * z[0][i] + sbeta[1] * z[1][i] + sbeta[2] * z[2][i];
    out[OUT0 + g * 256 + i] = o;
    out[OUT2 + g * 256 + i] = o;
    out[EMB1 + g * 256 + i] = e1[g * 256 + i];
    out[COM1 + g * 256 + i] = c1[g * 256 + i];
    out[COM2 + g * 256 + i] = c2[g * 256 + i];
    out[EMB2 + g * 256 + i] = e2[g * 256 + i];
  }
  if (t < 3) out[BETA + g * 3 + t] = sbeta[t];
}

// ---------------- host-side branch driver ----------------
static void run_branch(const float* x, const int* src0, const int* dst0,
                       const float* const* P, float* emb,
                       float* A, float* Bf, float* C,
                       float* scoreBuf, int* perm, int* mapping, float* keptF,
                       int* esrc, int* edst, float* emask, int E, hipStream_t stream) {
  const int M0 = NB * 512, M1 = NB * 256, M2 = NB * 128;
  const int EB8 = (E + 7) / 8, EB256 = (E + 255) / 256;
  hipMemsetAsync(emb, 0, NB * 256 * sizeof(float), stream);
  // layer 1
  gemm_bias_f32<<<M0 / 64, 256, 0, stream>>>(x, P[0], P[1], A, M0);
  hipMemsetAsync(Bf, 0, (size_t)M0 * HDIM * sizeof(float), stream);
  edge_scatter<<<EB8, 256, 0, stream>>>(A, src0, dst0, nullptr, Bf, E);
  relu_inplace<<<(M0 * HDIM) / 256, 256, 0, stream>>>(Bf, M0 * HDIM);
  node_score<<<M0 / 8, 256, 0, stream>>>(Bf, P[6], scoreBuf, M0);
  hipMemsetAsync(mapping, 0, M0 * sizeof(int), stream);
  hipMemsetAsync(keptF, 0, M0 * sizeof(float), stream);
  topk_select<<<NB, 512, 0, stream>>>(scoreBuf, 512, 256, perm, mapping, keptF);
  gather_scale<<<M1 / 8, 256, 0, stream>>>(Bf, scoreBuf, perm, C, M1);
  edge_remap<<<EB256, 256, 0, stream>>>(src0, dst0, nullptr, mapping, keptF, esrc, edst, emask, E);
  readout_acc<<<NB, 128, 0, stream>>>(C, emb, 256);
  // layer 2
  gemm_bias_f32<<<M1 / 64, 256, 0, stream>>>(C, P[2], P[3], A, M1);
  hipMemsetAsync(Bf, 0, (size_t)M1 * HDIM * sizeof(float), stream);
  edge_scatter<<<EB8, 256, 0, stream>>>(A, esrc, edst, emask, Bf, E);
  relu_inplace<<<(M1 * HDIM) / 256, 256, 0, stream>>>(Bf, M1 * HDIM);
  node_score<<<M1 / 8, 256, 0, stream>>>(Bf, P[7], scoreBuf, M1);
  hipMemsetAsync(mapping, 0, M1 * sizeof(int), stream);
  hipMemsetAsync(keptF, 0, M1 * sizeof(float), stream);
  topk_select<<<NB, 256, 0, stream>>>(scoreBuf, 256, 128, perm, mapping, keptF);
  gather_scale<<<M2 / 8, 256, 0, stream>>>(Bf, scoreBuf, perm, C, M2);
  edge_remap<<<EB256, 256, 0, stream>>>(esrc, edst, emask, mapping, keptF, esrc, edst, emask, E);
  readout_acc<<<NB, 128, 0, stream>>>(C, emb, 128);
  // layer 3
  gemm_bias_f32<<<M2 / 64, 256, 0, stream>>>(C, P[4], P[5], A, M2);
  hipMemsetAsync(Bf, 0, (size_t)M2 * HDIM * sizeof(float), stream);
  edge_scatter<<<EB8, 256, 0, stream>>>(A, esrc, edst, emask, Bf, E);
  relu_inplace<<<(M2 * HDIM) / 256, 256, 0, stream>>>(Bf, M2 * HDIM);
  readout_acc<<<NB, 128, 0, stream>>>(Bf, emb, 128);
}

extern "C" void kernel_launch(void* const* d_in, const int* in_sizes, int n_in,
                              void* d_out, int out_size, void* d_ws, size_t ws_size,
                              hipStream_t stream) {
  const float* x   = (const float*)d_in[0];
  const int* sadj  = (const int*)d_in[1];
  const int* fadj  = (const int*)d_in[2];
  const int E = in_sizes[1] / 2;

  const float* sg1[8]; const float* sg2[8]; const float* cg[8];
  for (int i = 0; i < 8; ++i) {
    sg1[i] = (const float*)d_in[4 + i];
    sg2[i] = (const float*)d_in[12 + i];
    cg[i]  = (const float*)d_in[20 + i];
  }
  const float* attW1 = (const float*)d_in[28];
  const float* attb1 = (const float*)d_in[29];
  const float* attw2 = (const float*)d_in[30];

  // workspace carve-up (256B aligned)
  char* w = (char*)d_ws;
  size_t off = 0;
  auto carve = [&](size_t bytes) {
    void* p = w + off;
    off = (off + bytes + 255) & ~(size_t)255;
    return p;
  };
  const int M0 = NB * 512;
  float* A       = (float*)carve((size_t)M0 * HDIM * sizeof(float));   // 16 MB
  float* Bf      = (float*)carve((size_t)M0 * HDIM * sizeof(float));   // 16 MB
  float* C       = (float*)carve((size_t)(M0 / 2) * HDIM * sizeof(float)); // 8 MB
  float* scoreB  = (float*)carve((size_t)M0 * sizeof(float));
  int*   perm    = (int*)carve((size_t)(M0 / 2) * sizeof(int));
  int*   mapping = (int*)carve((size_t)M0 * sizeof(int));
  float* keptF   = (float*)carve((size_t)M0 * sizeof(float));
  int*   esrc    = (int*)carve((size_t)E * sizeof(int));
  int*   edst    = (int*)carve((size_t)E * sizeof(int));
  float* emask   = (float*)carve((size_t)E * sizeof(float));
  float* emb1    = (float*)carve((size_t)NB * 256 * sizeof(float));
  float* com1    = (float*)carve((size_t)NB * 256 * sizeof(float));
  float* com2    = (float*)carve((size_t)NB * 256 * sizeof(float));
  float* emb2    = (float*)carve((size_t)NB * 256 * sizeof(float));

  const int* ssrc = sadj;           const int* sdst = sadj + E;
  const int* fsrc = fadj;           const int* fdst = fadj + E;

  run_branch(x, ssrc, sdst, sg1, emb1, A, Bf, C, scoreB, perm, mapping, keptF,
             esrc, edst, emask, E, stream);
  run_branch(x, ssrc, sdst, cg,  com1, A, Bf, C, scoreB, perm, mapping, keptF,
             esrc, edst, emask, E, stream);
  run_branch(x, fsrc, fdst, cg,  com2, A, Bf, C, scoreB, perm, mapping, keptF,
             esrc, edst, emask, E, stream);
  run_branch(x, fsrc, fdst, sg2, emb2, A, Bf, C, scoreB, perm, mapping, keptF,
             esrc, edst, emask, E, stream);

  finalize_att<<<NB, 128, 0, stream>>>(emb1, com1, com2, emb2,
                                       attW1, attb1, attw2, (float*)d_out);
  (void)n_in; (void)out_size; (void)ws_size;
}